// PINN_83691732730138
// MI455X (gfx1250) — compile-verified
//
#include <hip/hip_runtime.h>
#include <math.h>

typedef float v2f __attribute__((ext_vector_type(2)));
typedef float v8f __attribute__((ext_vector_type(8)));

#define WAVES_PER_BLOCK 2
#define PTS_PER_WAVE 16
#define PTS_PER_BLOCK (WAVES_PER_BLOCK * PTS_PER_WAVE)

// Evaluate one scalar MLP (2->64->64->1, tanh) together with its 2-jet
// (value, d/dx, d/dy, d2/dx2, d2/dxdy, d2/dy2) for the 16 points owned by
// this wave.  Results (6 floats) are valid in lanes 0..15 (pt = lane).
__device__ __forceinline__ void eval_mlp_jet(
    const float* __restrict__ W1, const float* __restrict__ B1,
    const float* __restrict__ W2, const float* __restrict__ B2,
    const float* __restrict__ WF, const float* __restrict__ BF,
    float* __restrict__ H1,          // this wave's LDS slice: 6*64*16 floats
    float xi, float yi, int hi, int pt, float out6[6])
{
    __syncthreads();   // H1 slice may be reused from a previous MLP pass

    // ---------------- layer 1: 2 -> 64, trivial input jets ----------------
    // lane handles rows {2*i + hi}, column pt.
    #pragma unroll 4
    for (int i = 0; i < 32; ++i) {
        const int row = 2 * i + hi;
        const float w0 = W1[row * 2 + 0];
        const float w1 = W1[row * 2 + 1];
        const float z  = fmaf(w0, xi, fmaf(w1, yi, B1[row]));
        const float t  = tanhf(z);
        const float d  = 1.0f - t * t;      // tanh'
        const float dd = -2.0f * t * d;     // tanh''
        const int o = row * 16 + pt;
        H1[0 * 1024 + o] = t;               // value
        H1[1 * 1024 + o] = d * w0;          // d/dx
        H1[2 * 1024 + o] = d * w1;          // d/dy
        H1[3 * 1024 + o] = dd * w0 * w0;    // d2/dx2
        H1[4 * 1024 + o] = dd * w0 * w1;    // d2/dxdy
        H1[5 * 1024 + o] = dd * w1 * w1;    // d2/dy2
    }
    __syncthreads();

    // ------- layer 2 GEMM (V_WMMA_F32_16X16X4_F32) + fused jet epilogue -----
    float f[6] = {0.f, 0.f, 0.f, 0.f, 0.f, 0.f};   // partial wf . h2 per channel

    for (int mt = 0; mt < 4; ++mt) {               // M tiles of 16 rows
        const int arow = mt * 16 + pt;             // A: M = lane%16
        v8f acc[6] = {};                           // one accumulator per channel

        // unroll capped at 4: keeps A(8)+B(48)+acc(48) VGPRs < 256, still
        // 24 back-to-back WMMAs per chunk for latency hiding.
        #pragma unroll 4
        for (int k = 0; k < 16; ++k) {
            // A fragment: VGPR0/1 hold K = 4k+2*hi, 4k+2*hi+1 (ISA 16x4 layout)
            const v2f a = *(const v2f*)(W2 + arow * 64 + (4 * k + 2 * hi));
            const int kr = 4 * k + 2 * hi;         // B: K rows for this lane half
            #pragma unroll
            for (int c = 0; c < 6; ++c) {
                v2f b;
                b.x = H1[c * 1024 + kr * 16 + pt];
                b.y = H1[c * 1024 + (kr + 1) * 16 + pt];
                acc[c] = __builtin_amdgcn_wmma_f32_16x16x4_f32(
                    false, a, false, b, (short)0, acc[c], false, false);
            }
        }

        // epilogue: lane holds rows mt*16 + 8*hi + r, col pt, all 6 channels
        #pragma unroll
        for (int r = 0; r < 8; ++r) {
            const int row = mt * 16 + 8 * hi + r;
            const float z0  = acc[0][r] + B2[row];
            const float zx  = acc[1][r];
            const float zy  = acc[2][r];
            const float zxx = acc[3][r];
            const float zxy = acc[4][r];
            const float zyy = acc[5][r];
            const float t  = tanhf(z0);
            const float d  = 1.0f - t * t;
            const float dd = -2.0f * t * d;
            const float hx  = d * zx;
            const float hy  = d * zy;
            const float hxx = fmaf(dd, zx * zx, d * zxx);
            const float hxy = fmaf(dd, zx * zy, d * zxy);
            const float hyy = fmaf(dd, zy * zy, d * zyy);
            const float w = WF[row];
            f[0] = fmaf(w, t,   f[0]);
            f[1] = fmaf(w, hx,  f[1]);
            f[2] = fmaf(w, hy,  f[2]);
            f[3] = fmaf(w, hxx, f[3]);
            f[4] = fmaf(w, hxy, f[4]);
            f[5] = fmaf(w, hyy, f[5]);
        }
    }

    // combine the two lane-halves (complementary row sets, same pt)
    #pragma unroll
    for (int c = 0; c < 6; ++c) f[c] += __shfl_xor(f[c], 16, 32);

    // ---------------- final tanh jet ----------------
    const float zf = f[0] + BF[0];
    const float t  = tanhf(zf);
    const float d  = 1.0f - t * t;
    const float dd = -2.0f * t * d;
    out6[0] = t;
    out6[1] = d * f[1];
    out6[2] = d * f[2];
    out6[3] = fmaf(dd, f[1] * f[1], d * f[3]);
    out6[4] = fmaf(dd, f[1] * f[2], d * f[4]);
    out6[5] = fmaf(dd, f[2] * f[2], d * f[5]);
}

__global__ __launch_bounds__(PTS_PER_BLOCK * 2)
void pinn_jet_kernel(
    const float* __restrict__ X,  const float* __restrict__ Y,
    const float* __restrict__ Uw1, const float* __restrict__ Ub1,
    const float* __restrict__ Uw2, const float* __restrict__ Ub2,
    const float* __restrict__ Uwf, const float* __restrict__ Ubf,
    const float* __restrict__ Vw1, const float* __restrict__ Vb1,
    const float* __restrict__ Vw2, const float* __restrict__ Vb2,
    const float* __restrict__ Vwf, const float* __restrict__ Vbf,
    float* __restrict__ out, int N)
{
    __shared__ float H1s[WAVES_PER_BLOCK][6 * 64 * 16];   // 24KB per wave

    const int lane = threadIdx.x & 31;
    const int wave = threadIdx.x >> 5;
    const int hi   = lane >> 4;
    const int pt   = lane & 15;
    float* H1 = H1s[wave];

    const int base = (blockIdx.x * WAVES_PER_BLOCK + wave) * PTS_PER_WAVE;
    const float xi = X[base + pt];
    const float yi = Y[base + pt];

    float uj[6], vj[6];
    eval_mlp_jet(Uw1, Ub1, Uw2, Ub2, Uwf, Ubf, H1, xi, yi, hi, pt, uj);
    eval_mlp_jet(Vw1, Vb1, Vw2, Vb2, Vwf, Vbf, H1, xi, yi, hi, pt, vj);

    if (lane < 16) {
        const float u   = uj[0], ux  = uj[1], uy  = uj[2];
        const float uxx = uj[3], uxy = uj[4], uyy = uj[5];
        const float v   = vj[0], vx  = vj[1], vy  = vj[2];
        const float vxx = vj[3], vxy = vj[4], vyy = vj[5];

        const float LMBD = 1.0f, MU = 0.5f, Q = 4.0f;
        const float C11 = 2.0f * MU + LMBD;   // 2.0
        const float C12 = LMBD;               // 1.0
        const float C33 = 2.0f * MU;          // 1.0

        const float Sxx = ux * C11 + vy * C12;
        const float Syy = vy * C11 + ux * C12;
        const float Sxy = 0.5f * C33 * uy * vx;

        const float Sxx_x = uxx * C11 + vxy * C12;                 // J[0,0]
        const float Syy_y = vyy * C11 + uxy * C12;                 // J[1,1]
        const float Sxy_x = 0.5f * C33 * (uxy * vx + uy * vxx);    // J[2,0]
        const float Sxy_y = 0.5f * C33 * (uyy * vx + uy * vxy);    // J[2,1]

        const float pi  = 3.14159265358979323846f;
        const float pi2 = pi * pi;
        const float s2px = sinf(2.0f * pi * xi);
        const float c2px = cosf(2.0f * pi * xi);
        const float spx  = sinf(pi * xi);
        const float cpx  = cosf(pi * xi);
        const float spy  = sinf(pi * yi);
        const float cpy  = cosf(pi * yi);
        const float y2 = yi * yi, y3 = y2 * yi, y4 = y2 * y2;

        const float Fx = -LMBD * (4.0f * pi2 * c2px * spy - Q * y3 * pi * cpx)
                         - MU   * (pi2 * c2px * spy - Q * y3 * pi * cpx)
                         - 8.0f * MU * pi2 * c2px * spy;
        const float Fy =  LMBD * (3.0f * Q * y2 * spx - 2.0f * pi2 * cpy * s2px)
                         - MU   * (2.0f * pi2 * cpy * s2px + Q * y4 * pi2 * spx * 0.25f)
                         + 6.0f * Q * MU * y2 * spx;

        const float Lx = Sxx_x + Sxy_y - Fx;
        const float Ly = Sxy_x + Syy_y - Fy;

        const float ml = (xi == 0.0f) ? 1.0f : 0.0f;
        const float mr = (xi == 1.0f) ? 1.0f : 0.0f;
        const float mb = (yi == 0.0f) ? 1.0f : 0.0f;
        const float mt = (yi == 1.0f) ? 1.0f : 0.0f;

        const int gid = base + pt;
        out[0  * N + gid] = Lx;
        out[1  * N + gid] = Ly;
        out[2  * N + gid] = Sxx;
        out[3  * N + gid] = Syy;
        out[4  * N + gid] = Sxy;
        out[5  * N + gid] = ml * v;
        out[6  * N + gid] = ml * Sxx;
        out[7  * N + gid] = mr * v;
        out[8  * N + gid] = mr * Sxx;
        out[9  * N + gid] = mb * u;
        out[10 * N + gid] = mb * v;
        out[11 * N + gid] = mt * u;
        out[12 * N + gid] = mt * (Syy - (LMBD + 2.0f * MU) * Q * spx);
    }
}

extern "C" void kernel_launch(void* const* d_in, const int* in_sizes, int n_in,
                              void* d_out, int out_size, void* d_ws, size_t ws_size,
                              hipStream_t stream) {
    const float* X   = (const float*)d_in[0];
    const float* Y   = (const float*)d_in[1];
    const float* Uw1 = (const float*)d_in[2];
    const float* Ub1 = (const float*)d_in[3];
    const float* Uw2 = (const float*)d_in[4];
    const float* Ub2 = (const float*)d_in[5];
    const float* Uwf = (const float*)d_in[6];
    const float* Ubf = (const float*)d_in[7];
    const float* Vw1 = (const float*)d_in[8];
    const float* Vb1 = (const float*)d_in[9];
    const float* Vw2 = (const float*)d_in[10];
    const float* Vb2 = (const float*)d_in[11];
    const float* Vwf = (const float*)d_in[12];
    const float* Vbf = (const float*)d_in[13];
    float* out = (float*)d_out;

    const int N = in_sizes[0];                 // 524288
    const int blocks = (N + PTS_PER_BLOCK - 1) / PTS_PER_BLOCK;
    pinn_jet_kernel<<<blocks, PTS_PER_BLOCK * 2, 0, stream>>>(
        X, Y, Uw1, Ub1, Uw2, Ub2, Uwf, Ubf,
        Vw1, Vb1, Vw2, Vb2, Vwf, Vbf, out, N);
}